// CATCallerEncoderLayer_86449101734002
// MI455X (gfx1250) — compile-verified
//
#include <hip/hip_runtime.h>
#include <math.h>

// Problem constants (from reference)
#define TT   1024
#define BB   16
#define CC   512
#define HH   8
#define KK   31
#define PP   15
#define MROWS (TT * BB)      // 16384
#define N2    (2 * CC)       // 1024 (w1 width)
#define HK    (HH * KK)      // 248
#define HKPAD 256
#define KP    (CC / 8)       // 64 k-step-pairs (each pair = 2 WMMA k-steps of 4)

typedef __attribute__((ext_vector_type(2))) float v2f;
typedef __attribute__((ext_vector_type(4))) float v4f;
typedef __attribute__((ext_vector_type(8))) float v8f;

// ---------------------------------------------------------------------------
// Weight packer: rearrange W (512 x ncols, row-major, ld=ldw) into WMMA
// B-fragment order. Layout: P[((nt*KP + kp)*32 + lane)*4 + j]
//   lane -> (hf = lane>>4, lrow = lane&15), col = nt*16 + lrow
//   j    -> k-row = kp*8 + (j>>1)*4 + 2*hf + (j&1)
// After packing, one b128 per lane supplies the B fragments for TWO
// consecutive 16x16x4 WMMA k-steps, fully coalesced across lanes.
// ---------------------------------------------------------------------------
__global__ __launch_bounds__(256)
void pack_b_kernel(const float* __restrict__ W, float* __restrict__ P,
                   int ldw, int ncols, int total) {
    const int idx = blockIdx.x * 256 + threadIdx.x;
    if (idx >= total) return;
    const int j    = idx & 3;
    const int lane = (idx >> 2) & 31;
    const int kp   = (idx >> 7) & (KP - 1);
    const int nt   = idx >> 13;
    const int hf   = lane >> 4;
    const int lrow = lane & 15;
    const int col  = nt * 16 + lrow;
    const int kr   = kp * 8 + (j >> 1) * 4 + 2 * hf + (j & 1);
    P[idx] = (col < ncols) ? W[(size_t)kr * ldw + col] : 0.0f;
}

// ---------------------------------------------------------------------------
// Kernel 1: h = x @ w1 + b1 ; xg = h[:, :C] * sigmoid(h[:, C:])
// One wave per 16(M) x 32(N) xg tile: 4 accumulators (a0,a1,g0,g1) share
// each A fragment. Inner loop: 2 x b64 (A) + 4 x b128 (packed B) + 8 WMMA.
// ---------------------------------------------------------------------------
__global__ __launch_bounds__(32)
void gemm1_glu_kernel(const float* __restrict__ x,
                      const float* __restrict__ w1p,
                      const float* __restrict__ b1,
                      float* __restrict__ xg) {
    const int mt   = blockIdx.x;           // 0..MROWS/16-1
    const int nq   = blockIdx.y;           // 0..15 (32-col groups of a-half)
    const int lane = threadIdx.x & 31;
    const int lrow = lane & 15;
    const int hf   = lane >> 4;
    const int m0   = mt * 16;
    const int n0   = nq * 32;
    const int ntA  = n0 >> 4;              // a-half tiles: ntA, ntA+1
    const int ntG  = 32 + ntA;             // g-half tiles: ntG, ntG+1

    const float* arow = x + (size_t)(m0 + lrow) * CC + 2 * hf;
    const float* pA0 = w1p + ((size_t)(ntA    ) * KP) * 128 + lane * 4;
    const float* pA1 = w1p + ((size_t)(ntA + 1) * KP) * 128 + lane * 4;
    const float* pG0 = w1p + ((size_t)(ntG    ) * KP) * 128 + lane * 4;
    const float* pG1 = w1p + ((size_t)(ntG + 1) * KP) * 128 + lane * 4;

    v8f aA0 = {}, aA1 = {}, aG0 = {}, aG1 = {};

    #pragma unroll 2
    for (int kp = 0; kp < KP; ++kp) {
        v2f a0 = *(const v2f*)(arow + kp * 8);
        v2f a1 = *(const v2f*)(arow + kp * 8 + 4);
        v4f fA0 = *(const v4f*)(pA0 + kp * 128);
        v4f fA1 = *(const v4f*)(pA1 + kp * 128);
        v4f fG0 = *(const v4f*)(pG0 + kp * 128);
        v4f fG1 = *(const v4f*)(pG1 + kp * 128);
        v2f b;
        b.x = fA0.x; b.y = fA0.y;
        aA0 = __builtin_amdgcn_wmma_f32_16x16x4_f32(false, a0, false, b, (short)0, aA0, false, false);
        b.x = fA0.z; b.y = fA0.w;
        aA0 = __builtin_amdgcn_wmma_f32_16x16x4_f32(false, a1, false, b, (short)0, aA0, false, false);
        b.x = fA1.x; b.y = fA1.y;
        aA1 = __builtin_amdgcn_wmma_f32_16x16x4_f32(false, a0, false, b, (short)0, aA1, false, false);
        b.x = fA1.z; b.y = fA1.w;
        aA1 = __builtin_amdgcn_wmma_f32_16x16x4_f32(false, a1, false, b, (short)0, aA1, false, false);
        b.x = fG0.x; b.y = fG0.y;
        aG0 = __builtin_amdgcn_wmma_f32_16x16x4_f32(false, a0, false, b, (short)0, aG0, false, false);
        b.x = fG0.z; b.y = fG0.w;
        aG0 = __builtin_amdgcn_wmma_f32_16x16x4_f32(false, a1, false, b, (short)0, aG0, false, false);
        b.x = fG1.x; b.y = fG1.y;
        aG1 = __builtin_amdgcn_wmma_f32_16x16x4_f32(false, a0, false, b, (short)0, aG1, false, false);
        b.x = fG1.z; b.y = fG1.w;
        aG1 = __builtin_amdgcn_wmma_f32_16x16x4_f32(false, a1, false, b, (short)0, aG1, false, false);
    }

    const int colA0 = n0 + lrow;
    const int colA1 = n0 + 16 + lrow;
    const float bA0 = b1[colA0],      bA1 = b1[colA1];
    const float bG0 = b1[CC + colA0], bG1 = b1[CC + colA1];
    #pragma unroll
    for (int v = 0; v < 8; ++v) {
        const int row = m0 + v + 8 * hf;
        const float av0 = aA0[v] + bA0;
        const float gv0 = aG0[v] + bG0;
        const float av1 = aA1[v] + bA1;
        const float gv1 = aG1[v] + bG1;
        xg[(size_t)row * CC + colA0] = av0 * (1.0f / (1.0f + expf(-gv0)));
        xg[(size_t)row * CC + colA1] = av1 * (1.0f / (1.0f + expf(-gv1)));
    }
}

// ---------------------------------------------------------------------------
// Kernel 2: logits = xg @ wl_w + wl_b -> padded (MROWS x 256) buffer.
// One wave per 16x32 tile, 2 accumulators. Pad columns carry zeros from the
// packer, so no guards needed in the hot loop.
// ---------------------------------------------------------------------------
__global__ __launch_bounds__(32)
void gemm_wl_kernel(const float* __restrict__ xg,
                    const float* __restrict__ wlwp,
                    const float* __restrict__ wlb,
                    float* __restrict__ wlog) {
    const int mt   = blockIdx.x;           // 0..MROWS/16-1
    const int nq   = blockIdx.y;           // 0..7
    const int lane = threadIdx.x & 31;
    const int lrow = lane & 15;
    const int hf   = lane >> 4;
    const int m0   = mt * 16;
    const int n0   = nq * 32;
    const int nt   = n0 >> 4;

    const float* arow = xg + (size_t)(m0 + lrow) * CC + 2 * hf;
    const float* p0 = wlwp + ((size_t)(nt    ) * KP) * 128 + lane * 4;
    const float* p1 = wlwp + ((size_t)(nt + 1) * KP) * 128 + lane * 4;

    v8f acc0 = {}, acc1 = {};
    #pragma unroll 2
    for (int kp = 0; kp < KP; ++kp) {
        v2f a0 = *(const v2f*)(arow + kp * 8);
        v2f a1 = *(const v2f*)(arow + kp * 8 + 4);
        v4f f0 = *(const v4f*)(p0 + kp * 128);
        v4f f1 = *(const v4f*)(p1 + kp * 128);
        v2f b;
        b.x = f0.x; b.y = f0.y;
        acc0 = __builtin_amdgcn_wmma_f32_16x16x4_f32(false, a0, false, b, (short)0, acc0, false, false);
        b.x = f0.z; b.y = f0.w;
        acc0 = __builtin_amdgcn_wmma_f32_16x16x4_f32(false, a1, false, b, (short)0, acc0, false, false);
        b.x = f1.x; b.y = f1.y;
        acc1 = __builtin_amdgcn_wmma_f32_16x16x4_f32(false, a0, false, b, (short)0, acc1, false, false);
        b.x = f1.z; b.y = f1.w;
        acc1 = __builtin_amdgcn_wmma_f32_16x16x4_f32(false, a1, false, b, (short)0, acc1, false, false);
    }

    const int col0 = n0 + lrow;
    const int col1 = n0 + 16 + lrow;
    const float bias0 = (col0 < HK) ? wlb[col0] : 0.0f;
    const float bias1 = (col1 < HK) ? wlb[col1] : 0.0f;
    #pragma unroll
    for (int v = 0; v < 8; ++v) {
        const int row = m0 + v + 8 * hf;
        wlog[(size_t)row * HKPAD + col0] = acc0[v] + bias0;
        wlog[(size_t)row * HKPAD + col1] = acc1[v] + bias1;
    }
}

// ---------------------------------------------------------------------------
// Kernel 3: in-place softmax over the 31 taps of each (row, head).
// ---------------------------------------------------------------------------
__global__ __launch_bounds__(256)
void softmax_taps_kernel(float* __restrict__ wlog) {
    const int idx = blockIdx.x * 256 + threadIdx.x;   // row*8 + head
    if (idx >= MROWS * HH) return;
    const int row = idx >> 3;
    const int h   = idx & 7;
    float* p = wlog + (size_t)row * HKPAD + h * KK;

    float vals[KK];
    float m = -INFINITY;
    #pragma unroll
    for (int k = 0; k < KK; ++k) { vals[k] = p[k]; m = fmaxf(m, vals[k]); }
    float s = 0.0f;
    #pragma unroll
    for (int k = 0; k < KK; ++k) { vals[k] = expf(vals[k] - m); s += vals[k]; }
    const float inv = 1.0f / s;
    #pragma unroll
    for (int k = 0; k < KK; ++k) p[k] = vals[k] * inv;
}

// ---------------------------------------------------------------------------
// Kernel 4: dynamic conv along time.
// y[t,b,c] = sum_k xg[t+k-P, b, c] * w[t,b,c/64, k]
// xg (32MB) is L2-resident, so the 31x re-read stays on-chip.
// ---------------------------------------------------------------------------
__global__ __launch_bounds__(256)
void dynconv_kernel(const float* __restrict__ xg,
                    const float* __restrict__ wlog,
                    float* __restrict__ y) {
    __shared__ float ws[4 * KK];
    const int bid   = blockIdx.x;
    const int row   = bid >> 1;            // t*B + b
    const int chalf = bid & 1;
    const int tid   = threadIdx.x;
    const int c     = chalf * 256 + tid;
    const int t     = row >> 4;            // row / B

    if (tid < 4 * KK) {
        const int hl = tid / KK;
        const int k  = tid - hl * KK;
        ws[tid] = wlog[(size_t)row * HKPAD + (chalf * 4 + hl) * KK + k];
    }
    __syncthreads();

    const float* taps = ws + (tid >> 6) * KK;   // local head 0..3
    float acc = 0.0f;
    #pragma unroll
    for (int k = 0; k < KK; ++k) {
        const int tt = t + k - PP;
        if (tt >= 0 && tt < TT) {
            acc = fmaf(xg[((size_t)row + (size_t)(k - PP) * BB) * CC + c],
                       taps[k], acc);
        }
    }
    y[(size_t)row * CC + c] = acc;
}

// ---------------------------------------------------------------------------
// Kernel 5: out = y @ w2 + b2. One wave per 16x64 tile, 4 accumulators.
// ---------------------------------------------------------------------------
__global__ __launch_bounds__(32)
void gemm2_kernel(const float* __restrict__ y,
                  const float* __restrict__ w2p,
                  const float* __restrict__ b2,
                  float* __restrict__ out) {
    const int mt   = blockIdx.x;           // 0..MROWS/16-1
    const int nq   = blockIdx.y;           // 0..7 (64-col groups)
    const int lane = threadIdx.x & 31;
    const int lrow = lane & 15;
    const int hf   = lane >> 4;
    const int m0   = mt * 16;
    const int n0   = nq * 64;
    const int nt   = n0 >> 4;

    const float* arow = y + (size_t)(m0 + lrow) * CC + 2 * hf;
    const float* p0 = w2p + ((size_t)(nt    ) * KP) * 128 + lane * 4;
    const float* p1 = w2p + ((size_t)(nt + 1) * KP) * 128 + lane * 4;
    const float* p2 = w2p + ((size_t)(nt + 2) * KP) * 128 + lane * 4;
    const float* p3 = w2p + ((size_t)(nt + 3) * KP) * 128 + lane * 4;

    v8f acc0 = {}, acc1 = {}, acc2 = {}, acc3 = {};
    #pragma unroll 2
    for (int kp = 0; kp < KP; ++kp) {
        v2f a0 = *(const v2f*)(arow + kp * 8);
        v2f a1 = *(const v2f*)(arow + kp * 8 + 4);
        v4f f0 = *(const v4f*)(p0 + kp * 128);
        v4f f1 = *(const v4f*)(p1 + kp * 128);
        v4f f2 = *(const v4f*)(p2 + kp * 128);
        v4f f3 = *(const v4f*)(p3 + kp * 128);
        v2f b;
        b.x = f0.x; b.y = f0.y;
        acc0 = __builtin_amdgcn_wmma_f32_16x16x4_f32(false, a0, false, b, (short)0, acc0, false, false);
        b.x = f0.z; b.y = f0.w;
        acc0 = __builtin_amdgcn_wmma_f32_16x16x4_f32(false, a1, false, b, (short)0, acc0, false, false);
        b.x = f1.x; b.y = f1.y;
        acc1 = __builtin_amdgcn_wmma_f32_16x16x4_f32(false, a0, false, b, (short)0, acc1, false, false);
        b.x = f1.z; b.y = f1.w;
        acc1 = __builtin_amdgcn_wmma_f32_16x16x4_f32(false, a1, false, b, (short)0, acc1, false, false);
        b.x = f2.x; b.y = f2.y;
        acc2 = __builtin_amdgcn_wmma_f32_16x16x4_f32(false, a0, false, b, (short)0, acc2, false, false);
        b.x = f2.z; b.y = f2.w;
        acc2 = __builtin_amdgcn_wmma_f32_16x16x4_f32(false, a1, false, b, (short)0, acc2, false, false);
        b.x = f3.x; b.y = f3.y;
        acc3 = __builtin_amdgcn_wmma_f32_16x16x4_f32(false, a0, false, b, (short)0, acc3, false, false);
        b.x = f3.z; b.y = f3.w;
        acc3 = __builtin_amdgcn_wmma_f32_16x16x4_f32(false, a1, false, b, (short)0, acc3, false, false);
    }

    const int col0 = n0 + lrow;
    const float bias0 = b2[col0];
    const float bias1 = b2[col0 + 16];
    const float bias2 = b2[col0 + 32];
    const float bias3 = b2[col0 + 48];
    #pragma unroll
    for (int v = 0; v < 8; ++v) {
        const int row = m0 + v + 8 * hf;
        float* o = out + (size_t)row * CC + col0;
        o[0]  = acc0[v] + bias0;
        o[16] = acc1[v] + bias1;
        o[32] = acc2[v] + bias2;
        o[48] = acc3[v] + bias3;
    }
}

// ---------------------------------------------------------------------------
extern "C" void kernel_launch(void* const* d_in, const int* in_sizes, int n_in,
                              void* d_out, int out_size, void* d_ws, size_t ws_size,
                              hipStream_t stream) {
    const float* x   = (const float*)d_in[0];   // (T,B,C)
    const float* w1  = (const float*)d_in[1];   // (C, 2C)
    const float* b1  = (const float*)d_in[2];   // (2C,)
    const float* wlw = (const float*)d_in[3];   // (C, H*K)
    const float* wlb = (const float*)d_in[4];   // (H*K,)
    const float* w2  = (const float*)d_in[5];   // (C, C)
    const float* b2  = (const float*)d_in[6];   // (C,)
    float* out = (float*)d_out;                 // (T,B,C)

    float* ws = (float*)d_ws;
    size_t off = 0;
    float* xg   = ws + off; off += (size_t)MROWS * CC;      // 32 MB
    float* wlog = ws + off; off += (size_t)MROWS * HKPAD;   // 16 MB
    float* yb   = ws + off; off += (size_t)MROWS * CC;      // 32 MB
    float* w1p  = ws + off; off += (size_t)(N2 / 16)    * KP * 128;  // 2 MB
    float* wlwp = ws + off; off += (size_t)(HKPAD / 16) * KP * 128;  // 0.5 MB
    float* w2p  = ws + off; off += (size_t)(CC / 16)    * KP * 128;  // 1 MB

    // One-time weight packs into WMMA B-fragment order (coalesced b128 loads).
    {
        const int tot1 = (N2 / 16)    * KP * 128;
        const int totL = (HKPAD / 16) * KP * 128;
        const int tot2 = (CC / 16)    * KP * 128;
        pack_b_kernel<<<(tot1 + 255) / 256, 256, 0, stream>>>(w1,  w1p,  N2, N2, tot1);
        pack_b_kernel<<<(totL + 255) / 256, 256, 0, stream>>>(wlw, wlwp, HK, HK, totL);
        pack_b_kernel<<<(tot2 + 255) / 256, 256, 0, stream>>>(w2,  w2p,  CC, CC, tot2);
    }

    gemm1_glu_kernel<<<dim3(MROWS / 16, CC / 32), 32, 0, stream>>>(x, w1p, b1, xg);
    gemm_wl_kernel<<<dim3(MROWS / 16, HKPAD / 32), 32, 0, stream>>>(xg, wlwp, wlb, wlog);
    softmax_taps_kernel<<<(MROWS * HH + 255) / 256, 256, 0, stream>>>(wlog);
    dynconv_kernel<<<MROWS * 2, 256, 0, stream>>>(xg, wlog, yb);
    gemm2_kernel<<<dim3(MROWS / 16, CC / 64), 32, 0, stream>>>(yb, w2p, b2, out);
}